// GenericRaymarcher_53730040872981
// MI455X (gfx1250) — compile-verified
//
#include <hip/hip_runtime.h>
#include <stdint.h>

#define RPB 128   // rays per block == blockDim.x
#define NP  64    // samples per ray
#define NF  3     // feature channels

// ---------------- CDNA5 async global<->LDS helpers (inline asm) ----------------
// Per CDNA5 ISA 10.2: for a generic (flat) pointer into LDS, addr[31:0] IS the
// LDS byte address, so truncating a generic __shared__ pointer to 32 bits gives
// the LDS-offset VGPR value these instructions want. No addrspace cast needed.

#if defined(__has_builtin)
# if __has_builtin(__builtin_amdgcn_s_wait_asynccnt)
#  define WAIT_ASYNC0() __builtin_amdgcn_s_wait_asynccnt(0)
# endif
#endif
#ifndef WAIT_ASYNC0
# define WAIT_ASYNC0() asm volatile("s_wait_asynccnt 0x0" ::: "memory")
#endif

__device__ __forceinline__ void async_load_b128(void* lds_dst, const void* g_src) {
  unsigned loff = (unsigned)(uintptr_t)lds_dst;     // LDS byte address
  asm volatile("global_load_async_to_lds_b128 %0, %1, off"
               :: "v"(loff), "v"(g_src) : "memory");
}

__device__ __forceinline__ void async_store_b128(void* g_dst, void* lds_src) {
  unsigned loff = (unsigned)(uintptr_t)lds_src;     // LDS byte address
  asm volatile("global_store_async_from_lds_b128 %0, %1, off"
               :: "v"(g_dst), "v"(loff) : "memory");
}

// ---------------- main kernel: 1 ray per lane, LDS-staged ----------------
// LDS layout is chunk-major: for array X, 16B chunk c of ray r lives at
// sX[c*RPB + r]. Async B128 loads are then 16B-aligned and, across a wave
// (lane = ray), the *global* addresses of one staging instruction are 32
// consecutive 16B chunks -> fully coalesced 512B bursts.
__global__ __launch_bounds__(RPB) void raymarch_main(
    const float* __restrict__ dens, const float* __restrict__ feat,
    const float* __restrict__ len,  const float* __restrict__ bg,
    float* __restrict__ out_feat, float* __restrict__ out_depth,
    float* __restrict__ out_opac, float* __restrict__ out_w)
{
  __shared__ float4 s_dens[16 * RPB];  // 32 KB   (reused for weights)
  __shared__ float4 s_len [16 * RPB];  // 32 KB
  __shared__ float4 s_feat[48 * RPB];  // 96 KB

  const int t = threadIdx.x;
  const long long blockRay = (long long)blockIdx.x * RPB;

  const float* gd = dens + blockRay * NP;
  const float* gl = len  + blockRay * NP;
  const float* gf = feat + blockRay * (NP * NF);

  // ---- stage: global -> LDS (async, coalesced) ----
#pragma unroll
  for (int k = 0; k < 16; ++k) {            // densities: 16 chunks/ray
    int i = t + k * RPB;                    // global 16B-chunk index in block
    async_load_b128(&s_dens[(i & 15) * RPB + (i >> 4)], gd + i * 4);
  }
#pragma unroll
  for (int k = 0; k < 16; ++k) {            // lengths
    int i = t + k * RPB;
    async_load_b128(&s_len[(i & 15) * RPB + (i >> 4)], gl + i * 4);
  }
#pragma unroll
  for (int k = 0; k < 48; ++k) {            // features: 48 chunks/ray
    int i = t + k * RPB;
    int ray = i / 48, c = i - ray * 48;
    async_load_b128(&s_feat[c * RPB + ray], gf + i * 4);
  }
  WAIT_ASYNC0();
  __syncthreads();

  // per-lane views into LDS (float index = chunk*512 + t*4 + sub)
  const float* Dp = (const float*)s_dens + t * 4;
  const float* Lp = (const float*)s_len  + t * 4;
  const float* Fp = (const float*)s_feat + t * 4;
  float*       Wp = (float*)s_dens + t * 4;   // overwrite densities with weights

  float T = 1.0f, depth = 0.0f, f0 = 0.0f, f1 = 0.0f, f2 = 0.0f;
  float len_p = Lp[0];
#pragma unroll
  for (int p = 0; p < NP; ++p) {
    const int pn = p + 1;
    float len_n = (p < NP - 1) ? Lp[(pn >> 2) * 512 + (pn & 3)] : len_p; // delta=0 at tail
    float d  = fmaxf(Dp[(p >> 2) * 512 + (p & 3)], 0.0f);
    float e  = __expf(-(len_n - len_p) * d);   // exp(-delta*dens)
    float w  = (1.0f - e) * T;                 // capped * absorption_shifted
    T *= e;                                    // running transmittance
    const int e0 = 3 * p;
    f0 += w * Fp[((e0    ) >> 2) * 512 + ((e0    ) & 3)];
    f1 += w * Fp[((e0 + 1) >> 2) * 512 + ((e0 + 1) & 3)];
    f2 += w * Fp[((e0 + 2) >> 2) * 512 + ((e0 + 2) & 3)];
    depth += w * len_p;
    Wp[(p >> 2) * 512 + (p & 3)] = w;          // stash weight (read-before-write same slot)
    len_p = len_n;
  }

  const float opac = 1.0f - T;
  const long long ray = blockRay + t;
  const float b0 = bg[0], b1 = bg[1], b2 = bg[2];
  out_feat[ray * 3 + 0] = opac * f0 + (1.0f - opac) * b0;
  out_feat[ray * 3 + 1] = opac * f1 + (1.0f - opac) * b1;
  out_feat[ray * 3 + 2] = opac * f2 + (1.0f - opac) * b2;
  out_depth[ray] = depth;
  out_opac[ray] = opac;

  // ---- drain: weights LDS -> global (async, coalesced) ----
  __syncthreads();
  float* gw = out_w + blockRay * NP;
#pragma unroll
  for (int k = 0; k < 16; ++k) {
    int i = t + k * RPB;
    async_store_b128(gw + i * 4, &s_dens[(i & 15) * RPB + (i >> 4)]);
  }
  WAIT_ASYNC0();
}

// ---------------- scalar tail kernel (rays % RPB != 0; unused for 262144) ----
__global__ void raymarch_tail(
    const float* __restrict__ dens, const float* __restrict__ feat,
    const float* __restrict__ len,  const float* __restrict__ bg,
    float* __restrict__ out_feat, float* __restrict__ out_depth,
    float* __restrict__ out_opac, float* __restrict__ out_w,
    long long start, long long rays)
{
  long long ray = start + (long long)blockIdx.x * blockDim.x + threadIdx.x;
  if (ray >= rays) return;
  const float* D = dens + ray * NP;
  const float* L = len  + ray * NP;
  const float* F = feat + ray * NP * NF;
  float T = 1.0f, depth = 0.0f, f0 = 0.0f, f1 = 0.0f, f2 = 0.0f;
  float lp = L[0];
  for (int p = 0; p < NP; ++p) {
    float ln = (p < NP - 1) ? L[p + 1] : lp;
    float d = fmaxf(D[p], 0.0f);
    float e = __expf(-(ln - lp) * d);
    float w = (1.0f - e) * T; T *= e;
    f0 += w * F[3 * p]; f1 += w * F[3 * p + 1]; f2 += w * F[3 * p + 2];
    depth += w * lp;
    out_w[ray * NP + p] = w;
    lp = ln;
  }
  float op = 1.0f - T;
  out_feat[ray * 3 + 0] = op * f0 + (1.0f - op) * bg[0];
  out_feat[ray * 3 + 1] = op * f1 + (1.0f - op) * bg[1];
  out_feat[ray * 3 + 2] = op * f2 + (1.0f - op) * bg[2];
  out_depth[ray] = depth;
  out_opac[ray] = op;
}

extern "C" void kernel_launch(void* const* d_in, const int* in_sizes, int n_in,
                              void* d_out, int out_size, void* d_ws, size_t ws_size,
                              hipStream_t stream) {
  (void)n_in; (void)out_size; (void)d_ws; (void)ws_size;
  const float* dens = (const float*)d_in[0];   // (B,R,P,1)
  const float* feat = (const float*)d_in[1];   // (B,R,P,3)
  const float* len  = (const float*)d_in[2];   // (B,R,P)
  const float* bg   = (const float*)d_in[3];   // (3,)

  const long long rays = (long long)in_sizes[0] / NP;   // B*R

  float* out       = (float*)d_out;
  float* out_feat  = out;                 // rays*3
  float* out_depth = out + rays * 3;      // rays
  float* out_opac  = out + rays * 4;      // rays
  float* out_w     = out + rays * 5;      // rays*64

  const long long fullBlocks = rays / RPB;
  if (fullBlocks > 0) {
    hipLaunchKernelGGL(raymarch_main, dim3((unsigned)fullBlocks), dim3(RPB), 0, stream,
                       dens, feat, len, bg, out_feat, out_depth, out_opac, out_w);
  }
  const long long rem = rays - fullBlocks * RPB;
  if (rem > 0) {
    unsigned tb = (unsigned)((rem + 127) / 128);
    hipLaunchKernelGGL(raymarch_tail, dim3(tb), dim3(128), 0, stream,
                       dens, feat, len, bg, out_feat, out_depth, out_opac, out_w,
                       fullBlocks * RPB, rays);
  }
}